// Embedding_61993557950654
// MI455X (gfx1250) — compile-verified
//
#include <hip/hip_runtime.h>
#include <stdint.h>

// 2-bit quantized embedding decode, MI455X (gfx1250, wave32).
//
// Mapping: one wave32 per token.
//   lane L handles dims 4L..4L+3  (128 dims / wave)
//   word  = bit_arr[token*8 + L/4]        (4 lanes share a word -> coalesced)
//   byte  = word >> (8*(L&3))             (8 bits = four 2-bit codes)
//   out[token*128 + 4L .. 4L+3] = codebook[code_i]   (128-bit NT store)
//
// Store-bandwidth bound: ~134 MB out + ~10 MB in => ~6.2 us floor @ 23.3 TB/s.
// NT stores keep the 12.8 MB bit_arr resident in the 192 MB L2 across replays.

typedef float v4f __attribute__((ext_vector_type(4)));

__global__ __launch_bounds__(256) void embed2b_decode(
    const int* __restrict__ tokens,        // [ntokens]
    const uint32_t* __restrict__ bit_arr,  // [ntokens_vocab*8] packed 2-bit codes
    const float* __restrict__ codebook,    // [4]
    float* __restrict__ out,               // [ntokens, 128]
    int ntokens)
{
    // --- Stage the 16-byte codebook into LDS via the CDNA5 async path ---
    __shared__ float lds_cb[4];
    if (threadIdx.x == 0) {
        uint32_t lds_off = (uint32_t)(uintptr_t)(&lds_cb[0]); // LDS offset (low 32b of generic addr)
        asm volatile("global_load_async_to_lds_b128 %0, %1, off"
                     :: "v"(lds_off), "v"(codebook)
                     : "memory");
        asm volatile("s_wait_asynccnt 0" ::: "memory");
    }
    __syncthreads();
    const float c0 = lds_cb[0];
    const float c1 = lds_cb[1];
    const float c2 = lds_cb[2];
    const float c3 = lds_cb[3];

    const int lane = threadIdx.x & 31;
    const int wid  = (int)((blockIdx.x * blockDim.x + threadIdx.x) >> 5); // wave id == token slot
    if (wid >= ntokens) return;

    // Token id (same address across the wave -> single coalesced request).
    const int tok = tokens[wid];

    // One packed word per group of 4 lanes (32B per token, coalesced gather; L2-resident).
    const uint32_t w    = bit_arr[(uint64_t)(uint32_t)tok * 8u + (uint32_t)(lane >> 2)];
    const uint32_t byte = (w >> ((lane & 3) * 8)) & 0xFFu;

    // Decode four 2-bit codes -> codebook values (3x v_cndmask each).
    v4f r;
#pragma unroll
    for (int i = 0; i < 4; ++i) {
        const uint32_t c = (byte >> (2 * i)) & 3u;
        const float a = (c & 1u) ? c1 : c0;
        const float b = (c & 1u) ? c3 : c2;
        r[i] = (c & 2u) ? b : a;
    }

    // 128-bit non-temporal store: wave writes one contiguous 512B output row.
    v4f* dst = (v4f*)(out + (uint64_t)wid * 128u + (uint32_t)(lane * 4));
    __builtin_nontemporal_store(r, dst);
}

extern "C" void kernel_launch(void* const* d_in, const int* in_sizes, int n_in,
                              void* d_out, int out_size, void* d_ws, size_t ws_size,
                              hipStream_t stream)
{
    const int*      tokens   = (const int*)d_in[0];
    const uint32_t* bit_arr  = (const uint32_t*)d_in[1];
    const float*    codebook = (const float*)d_in[2];
    float*          out      = (float*)d_out;

    const int ntokens = in_sizes[0];                 // 64*4096 = 262144
    const int threads = 256;                         // 8 waves / block
    const long long total_threads = (long long)ntokens * 32; // one wave32 per token
    const int blocks = (int)((total_threads + threads - 1) / threads);

    embed2b_decode<<<blocks, threads, 0, stream>>>(tokens, bit_arr, codebook, out, ntokens);
}